// MEGATLayer_81570018886031
// MI455X (gfx1250) — compile-verified
//
#include <hip/hip_runtime.h>
#include <hip/hip_bf16.h>
#include <math.h>

// MEGAT layer for MI455X (gfx1250). fp32 end-to-end (workload is HBM-bound:
// ~420MB traffic vs 6.4 GFLOP), matrix ops via V_WMMA_F32_16X16X4_F32.
// Single-use streams (e reads, alpha/out stores) use non-temporal hints so
// the reusable working set (h, x, W, adj) stays hot in the 192MB L2.
//
// Dims (fixed by reference): B=256, N=128, H=8, F=256, FH=32.

#define BB 256
#define NN 128
#define HH 8
#define FF 256
#define FH 32

typedef float v2f __attribute__((ext_vector_type(2)));
typedef float v8f __attribute__((ext_vector_type(8)));

__device__ __forceinline__ v8f wmma_f32_k4(v2f a, v2f b, v8f c) {
  // (neg_a, A, neg_b, B, c_mod, C, reuse_a, reuse_b)
  return __builtin_amdgcn_wmma_f32_16x16x4_f32(false, a, false, b, (short)0, c,
                                               false, false);
}

// ---------------------------------------------------------------------------
// Kernel 1: h = x @ W      x:[B,N,256]  W:[256,256]  h:[B,N,256] (workspace)
// Block = 256 thr (8 waves) handles one (b, 16-row panel); wave w owns col
// tiles 2w, 2w+1. A-panel staged in LDS (stride 268: 16B aligned rows, bank
// stride 12 -> 16 distinct banks for the 16-lane column reads).
// ---------------------------------------------------------------------------
#define XS_STRIDE 268

__global__ __launch_bounds__(256) void megat_proj_kernel(
    const float* __restrict__ x, const float* __restrict__ W,
    float* __restrict__ h) {
  __shared__ float xs[16 * XS_STRIDE];
  const int b = blockIdx.x >> 3;   // batch
  const int ti = blockIdx.x & 7;   // 16-row panel
  const int t = threadIdx.x;
  const int lane = t & 31;
  const int wave = t >> 5;

  // stage x panel [16 x 256] into LDS (1024 float4 / 256 threads)
  const float* xp = x + (size_t)(b * NN + ti * 16) * FF;
  #pragma unroll
  for (int q = 0; q < 4; ++q) {
    int f4 = t + 256 * q;          // 0..1023
    int row = f4 >> 6;             // 64 float4 per row
    int c4 = (f4 & 63) << 2;
    *(float4*)&xs[row * XS_STRIDE + c4] = *(const float4*)&xp[row * FF + c4];
  }
  __syncthreads();

  const int m = lane & 15;               // M / N index within tile
  const int koff = (lane >> 4) << 1;     // K sub-offset (0 or 2)
  const int tj0 = wave * 2;
  const float* w0 = W + tj0 * 16 + m;
  const float* w1 = W + tj0 * 16 + 16 + m;

  v8f c0 = {};
  v8f c1 = {};
  #pragma unroll 4
  for (int k = 0; k < FF; k += 4) {
    v2f a;
    a.x = xs[m * XS_STRIDE + k + koff];
    a.y = xs[m * XS_STRIDE + k + koff + 1];
    v2f b0;
    b0.x = w0[(size_t)(k + koff) * FF];
    b0.y = w0[(size_t)(k + koff + 1) * FF];
    v2f b1;
    b1.x = w1[(size_t)(k + koff) * FF];
    b1.y = w1[(size_t)(k + koff + 1) * FF];
    c0 = wmma_f32_k4(a, b0, c0);
    c1 = wmma_f32_k4(a, b1, c1);
  }

  float* hp = h + (size_t)(b * NN + ti * 16) * FF;
  const int rhalf = (lane >> 4) << 3;    // 0 or 8
  #pragma unroll
  for (int r = 0; r < 8; ++r) {
    int row = r + rhalf;
    hp[row * FF + tj0 * 16 + m] = c0[r];
    hp[row * FF + tj0 * 16 + 16 + m] = c1[r];
  }
}

// ---------------------------------------------------------------------------
// Kernel 2: attention + aggregation for one (b, head) per block.
//   scores = leaky_relu(el_i + er_j, .2) + e; mask adj>.5 -> -1e9; softmax_j
//   alpha -> global (returned, NT store) and LDS; out_head = alpha @ h_head;
//   out = elu(out_head + bias + x)
// Softmax/aggregation done in two 64-row halves so static LDS stays < 64KB.
// ---------------------------------------------------------------------------
#define HS_STRIDE 36   // 128x32 h-slice, 16B-aligned rows, conflict-free B reads
#define AS_STRIDE 132  // 64x128 alpha half, bank stride 4 -> conflict-free A reads

__global__ __launch_bounds__(256) void megat_attn_kernel(
    const float* __restrict__ adj, const float* __restrict__ e,
    const float* __restrict__ h, const float* __restrict__ x,
    const float* __restrict__ attn_l, const float* __restrict__ attn_r,
    const float* __restrict__ bias, float* __restrict__ out,
    float* __restrict__ alpha) {
  __shared__ float hs[NN * HS_STRIDE];        // 18.0 KB
  __shared__ float als[64 * AS_STRIDE];       // 33.0 KB
  __shared__ float elds[NN];
  __shared__ float erds[NN];

  const int bh = blockIdx.x;       // b*8 + head
  const int b = bh >> 3;
  const int head = bh & 7;
  const int t = threadIdx.x;
  const int lane = t & 31;
  const int wave = t >> 5;

  // 1. stage h[b, :, head*32 : +32] -> LDS (128x32, 1024 float4)
  const float* hp = h + (size_t)b * NN * FF + head * FH;
  #pragma unroll
  for (int q = 0; q < 4; ++q) {
    int f4 = t + 256 * q;          // 0..1023
    int row = f4 >> 3;             // 8 float4 per row
    int c4 = (f4 & 7) << 2;
    *(float4*)&hs[row * HS_STRIDE + c4] = *(const float4*)&hp[row * FF + c4];
  }
  __syncthreads();

  // 2. el / er (wave-uniform split: waves 0-3 -> er, waves 4-7 -> el)
  if (t < NN) {
    const float* ar = attn_r + head * FH;
    float s = 0.f;
    #pragma unroll
    for (int f = 0; f < FH; ++f) s += hs[t * HS_STRIDE + f] * ar[f];
    erds[t] = s;
  } else {
    int r = t - NN;
    const float* al = attn_l + head * FH;
    float s = 0.f;
    #pragma unroll
    for (int f = 0; f < FH; ++f) s += hs[r * HS_STRIDE + f] * al[f];
    elds[r] = s;
  }
  __syncthreads();

  const int m = lane & 15;
  const int koff = (lane >> 4) << 1;
  const int rhalf = (lane >> 4) << 3;
  float* arowbase = alpha + (size_t)bh * NN * NN;

  for (int half = 0; half < 2; ++half) {
    const int rbase = half * 64;

    // 3. softmax: each wave handles 8 rows of this half
    for (int rr = 0; rr < 8; ++rr) {
      int rl = wave * 8 + rr;       // local row in half (0..63)
      int row = rbase + rl;
      float eli = elds[row];
      const float* ep = e + ((size_t)bh * NN + row) * NN;
      const float* ap = adj + ((size_t)b * NN + row) * NN;
      float s[4];
      float mx = -3.4e38f;
      #pragma unroll
      for (int q = 0; q < 4; ++q) {
        int j = lane + 32 * q;
        float sc = eli + erds[j];
        sc = sc > 0.f ? sc : 0.2f * sc;                 // leaky_relu(.2)
        sc += __builtin_nontemporal_load(&ep[j]);       // e: single-use stream
        sc = (ap[j] > 0.5f) ? sc : -1e9f;               // threshold mask
        s[q] = sc;
        mx = fmaxf(mx, sc);
      }
      #pragma unroll
      for (int mm = 16; mm >= 1; mm >>= 1) mx = fmaxf(mx, __shfl_xor(mx, mm, 32));
      float sum = 0.f;
      #pragma unroll
      for (int q = 0; q < 4; ++q) {
        float ex = __expf(s[q] - mx);
        s[q] = ex;
        sum += ex;
      }
      #pragma unroll
      for (int mm = 16; mm >= 1; mm >>= 1) sum += __shfl_xor(sum, mm, 32);
      float inv = 1.0f / sum;
      float* arow = arowbase + (size_t)row * NN;
      #pragma unroll
      for (int q = 0; q < 4; ++q) {
        int j = lane + 32 * q;
        float av = s[q] * inv;
        als[rl * AS_STRIDE + j] = av;
        __builtin_nontemporal_store(av, &arow[j]);      // alpha: write-once
      }
    }
    __syncthreads();

    // 4. aggregation: out[i,f] = sum_j alpha[i,j]*hs[j,f]
    //    8 waves -> 4 local i-tiles x 2 f-tiles, K=128 -> 32 WMMA each
    const int i0l = (wave >> 1) * 16;     // local i-tile base in als
    const int f0 = (wave & 1) * 16;
    v8f c = {};
    #pragma unroll 4
    for (int k = 0; k < NN; k += 4) {
      v2f a;
      a.x = als[(i0l + m) * AS_STRIDE + k + koff];
      a.y = als[(i0l + m) * AS_STRIDE + k + koff + 1];
      v2f bb;
      bb.x = hs[(k + koff) * HS_STRIDE + f0 + m];
      bb.y = hs[(k + koff + 1) * HS_STRIDE + f0 + m];
      c = wmma_f32_k4(a, bb, c);
    }
    const int col = head * FH + f0 + m;
    const float bv = bias[col];
    #pragma unroll
    for (int r = 0; r < 8; ++r) {
      int row = rbase + i0l + r + rhalf;
      size_t off = ((size_t)b * NN + row) * FF + col;
      float v = c[r] + bv + x[off];            // bias + residual
      v = v > 0.f ? v : (__expf(v) - 1.0f);    // ELU
      __builtin_nontemporal_store(v, &out[off]);  // out: write-once
    }
    __syncthreads();  // als reused by next half
  }
}

// ---------------------------------------------------------------------------
// Inputs (setup_inputs order): adj, x, e, W, attn_l, attn_r, bias
// d_out = [ out: B*N*256 f32 | alpha: B*H*N*N f32 ]
// d_ws  = h: B*N*256 f32 (33.5 MB)
// ---------------------------------------------------------------------------
extern "C" void kernel_launch(void* const* d_in, const int* in_sizes, int n_in,
                              void* d_out, int out_size, void* d_ws,
                              size_t ws_size, hipStream_t stream) {
  const float* adj = (const float*)d_in[0];
  const float* x = (const float*)d_in[1];
  const float* e = (const float*)d_in[2];
  const float* W = (const float*)d_in[3];
  const float* attn_l = (const float*)d_in[4];
  const float* attn_r = (const float*)d_in[5];
  const float* bias = (const float*)d_in[6];

  float* out = (float*)d_out;
  float* alpha = out + (size_t)BB * NN * FF;
  float* h = (float*)d_ws;  // B*N*256 f32

  megat_proj_kernel<<<BB * (NN / 16), 256, 0, stream>>>(x, W, h);
  megat_attn_kernel<<<BB * HH, 256, 0, stream>>>(adj, e, h, x, attn_l, attn_r,
                                                 bias, out, alpha);
}